// InSituBackpropLayer_28441273434611
// MI455X (gfx1250) — compile-verified
//
#include <hip/hip_runtime.h>
#include <math.h>

// ---------------------------------------------------------------------------
// Problem constants (from reference)
// ---------------------------------------------------------------------------
#define N_WG   512          // rows
#define N_COLC 512          // mesh columns
#define MNODE  256          // nodes per column
#define NB     8192         // batch
#define PI4f   0.78539816339744830962f
#define LN10_20 0.11512925464970228420f   // ln(10)/20

struct cpx { float x, y; };
__device__ __forceinline__ cpx cmul(cpx a, cpx b) {
    return { a.x*b.x - a.y*b.y, a.x*b.y + a.y*b.x };
}
__device__ __forceinline__ cpx cadd(cpx a, cpx b) { return { a.x+b.x, a.y+b.y }; }

// ---------------------------------------------------------------------------
// Kernel 1: fully parallel — fold phase screens + both DC couplers of every
// node into one combined 2x2 complex matrix S = P * L * Q * R  (row-vector
// convention: [vt', vb'] = [vt, vb] @ S).  Removes all transcendentals from
// the serial mesh loop.
// ---------------------------------------------------------------------------
__global__ void node_mats_kernel(const float* __restrict__ thetas,
                                 const float* __restrict__ phis,
                                 const float* __restrict__ bs,    // [N,2]
                                 const float* __restrict__ loss,  // [N,2]
                                 float* __restrict__ Smat)        // [N,8]
{
    int n = blockIdx.x * blockDim.x + threadIdx.x;
    if (n >= N_COLC * MNODE) return;

    float th = thetas[n], ph = phis[n];
    float eL = bs[2*n + 0], eR = bs[2*n + 1];
    float insL = expf(LN10_20 * loss[2*n + 0]);
    float insR = expf(LN10_20 * loss[2*n + 1]);
    float sL = sinf(PI4f + eL), cL = cosf(PI4f + eL);
    float sR = sinf(PI4f + eR), cR = cosf(PI4f + eR);

    // DC matrices per reference: [[ins*s, i*ins*c], [i*c, s]]
    cpx L11{insL*sL, 0.f}, L12{0.f, insL*cL}, L21{0.f, cL}, L22{sL, 0.f};
    cpx R11{insR*sR, 0.f}, R12{0.f, insR*cR}, R21{0.f, cR}, R22{sR, 0.f};
    cpx ep{cosf(ph), sinf(ph)};   // e^{i phi}
    cpx et{cosf(th), sinf(th)};   // e^{i theta}

    // M = P @ L  (P = diag(e^{i phi}, 1) scales row 0)
    cpx M11 = cmul(ep, L11), M12 = cmul(ep, L12), M21 = L21, M22 = L22;
    // M = M @ Q  (Q = diag(e^{i theta}, 1) scales column 0)
    M11 = cmul(M11, et); M21 = cmul(M21, et);
    // S = M @ R
    cpx S11 = cadd(cmul(M11, R11), cmul(M12, R21));
    cpx S12 = cadd(cmul(M11, R12), cmul(M12, R22));
    cpx S21 = cadd(cmul(M21, R11), cmul(M22, R21));
    cpx S22 = cadd(cmul(M21, R12), cmul(M22, R22));

    float4* o = (float4*)(Smat + (size_t)n * 8);
    o[0] = make_float4(S11.x, S11.y, S12.x, S12.y);
    o[1] = make_float4(S21.x, S21.y, S22.x, S22.y);
}

// ---------------------------------------------------------------------------
// Kernel 2: serial mesh propagation of the identity -> builds U (512x512
// complex).  64 blocks, each owns UCOLS=8 columns of U in LDS (32 KB),
// 256 threads = 1 thread per node.  Inner loop is pure 2x2 complex FMA.
// ---------------------------------------------------------------------------
#define UCOLS 8

__global__ void __launch_bounds__(256)
mesh_prop_kernel(const float* __restrict__ gammas,
                 const int* __restrict__ top,
                 const int* __restrict__ bot,
                 const int* __restrict__ mask,
                 const float* __restrict__ Smat,
                 float* __restrict__ Ure,     // [512,512] row-major (m,k)
                 float* __restrict__ Uim)
{
    __shared__ float2 v[UCOLS][N_WG];   // 32 KB
    const int tid = threadIdx.x;        // 0..255 == node id
    const int c0  = blockIdx.x * UCOLS; // first U column handled here

    // init: v[:, j] = e_j * e^{i gamma_j}
    for (int i = tid; i < UCOLS * N_WG; i += 256) {
        int uc = i >> 9, r = i & (N_WG - 1);
        float2 val = make_float2(0.f, 0.f);
        if (r == c0 + uc) { float g = gammas[r]; val = make_float2(cosf(g), sinf(g)); }
        v[uc][r] = val;
    }
    __syncthreads();

    for (int c = 0; c < N_COLC; ++c) {
        const int n = c * MNODE + tid;
        if (mask[n]) {
            const int t = top[n], b = bot[n];
            const float4 s0 = ((const float4*)Smat)[2*n + 0];
            const float4 s1 = ((const float4*)Smat)[2*n + 1];
            const cpx s11{s0.x, s0.y}, s12{s0.z, s0.w};
            const cpx s21{s1.x, s1.y}, s22{s1.z, s1.w};
#pragma unroll
            for (int uc = 0; uc < UCOLS; ++uc) {
                float2 vt2 = v[uc][t], vb2 = v[uc][b];
                cpx vt{vt2.x, vt2.y}, vb{vb2.x, vb2.y};
                cpx nt = cadd(cmul(s11, vt), cmul(s21, vb));
                cpx nb = cadd(cmul(s12, vt), cmul(s22, vb));
                v[uc][t] = make_float2(nt.x, nt.y);
                v[uc][b] = make_float2(nb.x, nb.y);
            }
        }
        __syncthreads();
    }

    for (int i = tid; i < UCOLS * N_WG; i += 256) {
        int uc = i >> 9, r = i & (N_WG - 1);
        int j = c0 + uc;
        float2 val = v[uc][r];
        Ure[(size_t)r * N_WG + j] = val.x;
        Uim[(size_t)r * N_WG + j] = val.y;
    }
}

// ---------------------------------------------------------------------------
// Kernel 3: out.re = U.re @ x ; out.im = U.im @ x via V_WMMA_F32_16X16X4_F32.
// Block = 256 threads (8 waves).  All 8 waves in a block share one 64-wide
// N strip (B fragments hit WGP$/L2) and cover 8 consecutive 16-row M tiles.
// Each wave: 16x64 output strip, 4 v8f accumulators, K=512 in steps of 4.
// ---------------------------------------------------------------------------
typedef float v2f __attribute__((ext_vector_type(2)));
typedef float v8f __attribute__((ext_vector_type(8)));

__global__ void __launch_bounds__(256)
wmma_gemm_kernel(const float* __restrict__ Ure,
                 const float* __restrict__ Uim,
                 const float* __restrict__ X,    // [512, 8192]
                 float* __restrict__ Out)        // [2, 512, 8192]
{
    const int lane = threadIdx.x & 31;
    const int wv   = threadIdx.x >> 5;          // wave in block: 0..7
    const int l16  = lane & 15;
    const int hf   = lane >> 4;                 // 0 or 1 (lane half)

    // blockIdx.x in [0,1024): mat(2) x mtg(4) x nstrip(128)
    const int bid = blockIdx.x;
    const int mat = bid >> 9;                   // 0 = real, 1 = imag
    const int rem = bid & 511;
    const int mtg = rem >> 7;                   // 0..3
    const int nst = rem & 127;                  // 0..127

    const int m0 = (mtg * 8 + wv) * 16;         // 0..496
    const int n0 = nst * 64;

    const float* A = mat ? Uim : Ure;
    const float* arow = A + (size_t)(m0 + l16) * N_WG + 2 * hf;
    const float* brow = X + (size_t)(2 * hf) * NB + n0 + l16;

    v8f acc0 = {0,0,0,0,0,0,0,0};
    v8f acc1 = acc0, acc2 = acc0, acc3 = acc0;

#pragma unroll 2
    for (int k = 0; k < N_WG; k += 4) {
        // A fragment 16x4 f32: VGPR0 = {K=k+2hf}, VGPR1 = {K=k+2hf+1}
        const float2 av = *(const float2*)(arow + k);
        v2f a; a.x = av.x; a.y = av.y;

        const float* bk = brow + (size_t)k * NB;
        v2f b0; b0.x = bk[ 0];        b0.y = bk[ 0 + NB];
        v2f b1; b1.x = bk[16];        b1.y = bk[16 + NB];
        v2f b2; b2.x = bk[32];        b2.y = bk[32 + NB];
        v2f b3; b3.x = bk[48];        b3.y = bk[48 + NB];

        acc0 = __builtin_amdgcn_wmma_f32_16x16x4_f32(false, a, false, b0, (short)0, acc0, false, false);
        acc1 = __builtin_amdgcn_wmma_f32_16x16x4_f32(false, a, false, b1, (short)0, acc1, false, false);
        acc2 = __builtin_amdgcn_wmma_f32_16x16x4_f32(false, a, false, b2, (short)0, acc2, false, false);
        acc3 = __builtin_amdgcn_wmma_f32_16x16x4_f32(false, a, false, b3, (short)0, acc3, false, false);
    }

    float* O = Out + (size_t)mat * ((size_t)N_WG * NB);
#pragma unroll
    for (int r = 0; r < 8; ++r) {
        int row = m0 + 8 * hf + r;              // C/D layout: VGPR r, +8 for hi half
        float* orow = O + (size_t)row * NB + n0 + l16;
        orow[ 0] = acc0[r];
        orow[16] = acc1[r];
        orow[32] = acc2[r];
        orow[48] = acc3[r];
    }
}

// ---------------------------------------------------------------------------
// Launch.  Workspace layout (floats):
//   S    : 131072 * 8        = 1,048,576 floats (4 MB)
//   Ure  : 512 * 512         =   262,144 floats (1 MB)
//   Uim  : 512 * 512         =   262,144 floats (1 MB)
// ---------------------------------------------------------------------------
extern "C" void kernel_launch(void* const* d_in, const int* in_sizes, int n_in,
                              void* d_out, int out_size, void* d_ws, size_t ws_size,
                              hipStream_t stream)
{
    const float* x      = (const float*)d_in[0];
    const float* thetas = (const float*)d_in[1];
    const float* phis   = (const float*)d_in[2];
    const float* gammas = (const float*)d_in[3];
    const float* bs     = (const float*)d_in[4];
    const float* loss   = (const float*)d_in[5];
    const int*   top    = (const int*)d_in[6];
    const int*   bot    = (const int*)d_in[7];
    const int*   mask   = (const int*)d_in[8];

    float* ws  = (float*)d_ws;
    float* S   = ws;
    float* Ure = ws + (size_t)N_COLC * MNODE * 8;
    float* Uim = Ure + (size_t)N_WG * N_WG;
    float* out = (float*)d_out;

    node_mats_kernel<<<(N_COLC * MNODE) / 256, 256, 0, stream>>>(thetas, phis, bs, loss, S);
    mesh_prop_kernel<<<N_WG / UCOLS, 256, 0, stream>>>(gammas, top, bot, mask, S, Ure, Uim);
    wmma_gemm_kernel<<<1024, 256, 0, stream>>>(Ure, Uim, x, out);
}